// TransNeXt_64484638982150
// MI455X (gfx1250) — compile-verified
//
#include <hip/hip_runtime.h>

typedef __attribute__((ext_vector_type(2))) float v2f;
typedef __attribute__((ext_vector_type(8))) float v8f;

#define BN     16
#define CIN    256
#define NHEAD  8
#define CHD    8
#define HSZ    56
#define HWSZ   (HSZ*HSZ)
#define NQKV   192
#define NOUT   256
#define KPROJ  64
#define SCALE_F 0.17677669529663687f   // (256/8)^-0.5

// Async global->LDS (gfx1250): direct HBM/L2 -> LDS copy, ASYNCcnt-tracked,
// no VGPR round trip. Guarded so the kernel still builds (store fallback)
// on a toolchain without the builtin. Parameter types per hipcc diagnostic:
// (AS1 int*, AS3 int*, imm offset, imm cpol).
#if defined(__has_builtin)
#if __has_builtin(__builtin_amdgcn_global_load_async_to_lds_b32)
#define USE_ASYNC_LDS 1
#endif
#endif

typedef __attribute__((address_space(1))) int* gas_ip;
typedef __attribute__((address_space(3))) int* las_ip;

// ------------------------------------------------------------------
// Kernel 1: qkv GEMM.  Per batch b: M=3136 (spatial), K=256, N=192.
// A[m,k] = x[b][k][m]  (channel-planar -> M contiguous per k-plane)
// B[k,n] = qkv_w[n][k]
// Output f[b][n][m] planar (+bias), consumed by the attention kernel.
// One wave per 16x16 D tile; each wave owns 3 N-tiles so the A
// fragment is fetched once per K-step.
// ------------------------------------------------------------------
__global__ __launch_bounds__(128) void qkv_gemm_kernel(
    const float* __restrict__ x, const float* __restrict__ qkv_w,
    const float* __restrict__ qkv_b, float* __restrict__ f)
{
  const int b     = blockIdx.y;
  const int m0    = blockIdx.x << 4;
  const int lane  = threadIdx.x & 31;
  const int wave  = threadIdx.x >> 5;
  const int laneN = lane & 15;
  const int laneH = lane >> 4;     // 0: K=0,1 half, 1: K=2,3 half

  const float* xb = x + (size_t)b * CIN * HWSZ + m0 + laneN;

  v8f zacc = {};
  v8f acc[3];
  const float* wrow[3];
  int nbase[3];
#pragma unroll
  for (int i = 0; i < 3; ++i) {
    acc[i]   = zacc;
    nbase[i] = (wave + 4 * i) << 4;
    wrow[i]  = qkv_w + (size_t)(nbase[i] + laneN) * CIN;
  }

  for (int k0 = 0; k0 < CIN; k0 += 4) {
    const int ka = k0 + 2 * laneH;
    // speculative prefetch of the A-planes 4 K-steps ahead
    // (translation failures on the tail are silently dropped)
    __builtin_prefetch(xb + (size_t)(ka + 16) * HWSZ, 0, 0);
    v2f a;
    a.x = xb[(size_t)ka * HWSZ];
    a.y = xb[(size_t)(ka + 1) * HWSZ];
#pragma unroll
    for (int i = 0; i < 3; ++i) {
      v2f bv;
      bv.x = wrow[i][ka];
      bv.y = wrow[i][ka + 1];
      acc[i] = __builtin_amdgcn_wmma_f32_16x16x4_f32(
          false, a, false, bv, (short)0, acc[i], false, false);
    }
  }

#pragma unroll
  for (int i = 0; i < 3; ++i) {
    const int n     = nbase[i] + laneN;
    const float bia = qkv_b[n];
    float* dst = f + ((size_t)b * NQKV + n) * HWSZ + m0 + 8 * laneH;
#pragma unroll
    for (int r = 0; r < 8; ++r) dst[r] = acc[i][r] + bia;
  }
}

// ------------------------------------------------------------------
// Kernel 2: depthwise convs + 3x3-neighborhood attention.
// Both depthwise convs share kernels per output channel, so we
// pre-sum  wsum = dep_conv_w + dep_conv1_w,  cbs = dep_conv_b + dep_conv1_b
// (fully generic; the "identity" structure of dep_conv_w is not assumed).
// 16x16 spatial tile / block; 8 k-planes + 8 v-planes with 1-halo staged
// into LDS via async global->LDS copies (halo zeros via plain ds stores;
// disjoint addresses, so ASYNCcnt/DScnt ordering does not matter).
// ------------------------------------------------------------------
__global__ __launch_bounds__(256) void conv_attn_kernel(
    const float* __restrict__ f,
    const float* __restrict__ dcw, const float* __restrict__ dcb,
    const float* __restrict__ dc1w, const float* __restrict__ dc1b,
    const float* __restrict__ rpb,
    float* __restrict__ attn_out)
{
  __shared__ float sk[CHD][18][18];
  __shared__ float sv[CHD][18][18];
  __shared__ float swsum[CHD * 9 * 9];
  __shared__ float scbs[CHD * 9];
  __shared__ float srpb[9];

  const int tx  = threadIdx.x, ty = threadIdx.y;
  const int tid = ty * 16 + tx;
  const int b   = blockIdx.z >> 3;
  const int nh  = blockIdx.z & 7;
  const int x0  = blockIdx.x << 4;
  const int y0  = blockIdx.y << 4;

  for (int i = tid; i < CHD * 9 * 9; i += 256) swsum[i] = dcw[i] + dc1w[i];
  for (int i = tid; i < CHD * 9;     i += 256) scbs[i]  = dcb[i] + dc1b[i];
  if (tid < 9) srpb[tid] = rpb[nh * 9 + tid];

  const float* fb = f + ((size_t)b * NQKV + nh * 24) * HWSZ;
  for (int c = 0; c < CHD; ++c) {
    const float* kp = fb + (size_t)(8 + c)  * HWSZ;
    const float* vp = fb + (size_t)(16 + c) * HWSZ;
    for (int idx = tid; idx < 18 * 18; idx += 256) {
      const int ly = idx / 18, lx = idx % 18;
      const int gy = y0 + ly - 1, gx = x0 + lx - 1;
      const bool in = (gy >= 0) && (gy < HSZ) && (gx >= 0) && (gx < HSZ);
      const int g = gy * HSZ + gx;
#ifdef USE_ASYNC_LDS
      if (in) {
        __builtin_amdgcn_global_load_async_to_lds_b32(
            (gas_ip)(kp + g), (las_ip)&sk[c][ly][lx], 0, 0);
        __builtin_amdgcn_global_load_async_to_lds_b32(
            (gas_ip)(vp + g), (las_ip)&sv[c][ly][lx], 0, 0);
      } else {
        sk[c][ly][lx] = 0.0f;
        sv[c][ly][lx] = 0.0f;
      }
#else
      sk[c][ly][lx] = in ? kp[g] : 0.0f;
      sv[c][ly][lx] = in ? vp[g] : 0.0f;
#endif
    }
  }
#ifdef USE_ASYNC_LDS
  asm volatile("s_wait_asynccnt 0x0" ::: "memory");
#endif
  __syncthreads();

  const int hh = y0 + ty, ww = x0 + tx;
  if (hh >= HSZ || ww >= HSZ) return;
  const int pos = hh * HSZ + ww;

  float qv[CHD];
#pragma unroll
  for (int c = 0; c < CHD; ++c)
    qv[c] = fb[(size_t)c * HWSZ + pos] * SCALE_F;

  float logit[9];
#pragma unroll
  for (int p = 0; p < 9; ++p) logit[p] = 0.0f;

  // pass 1: k-values -> attention logits
#pragma unroll
  for (int c = 0; c < CHD; ++c) {
    float nb[9];
#pragma unroll
    for (int j = 0; j < 9; ++j) nb[j] = sk[c][ty + j / 3][tx + j % 3];
#pragma unroll
    for (int p = 0; p < 9; ++p) {
      const int o = c * 9 + p;
      float kvv = scbs[o] + srpb[p];
#pragma unroll
      for (int j = 0; j < 9; ++j) kvv += swsum[o * 9 + j] * nb[j];
      logit[p] += qv[c] * kvv;
    }
  }

  // softmax over 9 positions
  float mx = logit[0];
#pragma unroll
  for (int p = 1; p < 9; ++p) mx = fmaxf(mx, logit[p]);
  float wsm[9], ssum = 0.0f;
#pragma unroll
  for (int p = 0; p < 9; ++p) { wsm[p] = __expf(logit[p] - mx); ssum += wsm[p]; }
  const float inv = 1.0f / ssum;
#pragma unroll
  for (int p = 0; p < 9; ++p) wsm[p] *= inv;

  // pass 2: v-values -> weighted sum (recompute from LDS; register-light)
#pragma unroll
  for (int c = 0; c < CHD; ++c) {
    float nb[9];
#pragma unroll
    for (int j = 0; j < 9; ++j) nb[j] = sv[c][ty + j / 3][tx + j % 3];
    float oc = 0.0f;
#pragma unroll
    for (int p = 0; p < 9; ++p) {
      const int o = c * 9 + p;
      float vvv = scbs[o];
#pragma unroll
      for (int j = 0; j < 9; ++j) vvv += swsum[o * 9 + j] * nb[j];
      oc += wsm[p] * vvv;
    }
    attn_out[((size_t)b * 64 + nh * 8 + c) * HWSZ + pos] = oc;
  }
}

// ------------------------------------------------------------------
// Kernel 3: proj GEMM. Per b: M=3136, K=64, N=256.
// A[m,k] = attn[b][k][m];  B[k,n] = proj_w[n][k].
// d_out is channel-planar (B,C,H,W) flat, so D tiles store directly.
// ------------------------------------------------------------------
__global__ __launch_bounds__(128) void proj_gemm_kernel(
    const float* __restrict__ attn, const float* __restrict__ proj_w,
    const float* __restrict__ proj_b, float* __restrict__ out)
{
  const int b     = blockIdx.y;
  const int m0    = blockIdx.x << 4;
  const int lane  = threadIdx.x & 31;
  const int wave  = threadIdx.x >> 5;
  const int laneN = lane & 15;
  const int laneH = lane >> 4;

  const float* ab = attn + (size_t)b * KPROJ * HWSZ + m0 + laneN;

  v8f zacc = {};
  v8f acc[4];
  const float* wrow[4];
  int nbase[4];
#pragma unroll
  for (int i = 0; i < 4; ++i) {
    acc[i]   = zacc;
    nbase[i] = (wave + 4 * i) << 4;
    wrow[i]  = proj_w + (size_t)(nbase[i] + laneN) * KPROJ;
  }

  for (int k0 = 0; k0 < KPROJ; k0 += 4) {
    const int ka = k0 + 2 * laneH;
    __builtin_prefetch(ab + (size_t)(ka + 16) * HWSZ, 0, 0);
    v2f a;
    a.x = ab[(size_t)ka * HWSZ];
    a.y = ab[(size_t)(ka + 1) * HWSZ];
#pragma unroll
    for (int i = 0; i < 4; ++i) {
      v2f bv;
      bv.x = wrow[i][ka];
      bv.y = wrow[i][ka + 1];
      acc[i] = __builtin_amdgcn_wmma_f32_16x16x4_f32(
          false, a, false, bv, (short)0, acc[i], false, false);
    }
  }

#pragma unroll
  for (int i = 0; i < 4; ++i) {
    const int n     = nbase[i] + laneN;
    const float bia = proj_b[n];
    float* dst = out + ((size_t)b * NOUT + n) * HWSZ + m0 + 8 * laneH;
#pragma unroll
    for (int r = 0; r < 8; ++r) dst[r] = acc[i][r] + bia;
  }
}

// ------------------------------------------------------------------
extern "C" void kernel_launch(void* const* d_in, const int* in_sizes, int n_in,
                              void* d_out, int out_size, void* d_ws, size_t ws_size,
                              hipStream_t stream) {
  const float* x      = (const float*)d_in[0];
  const float* qkv_w  = (const float*)d_in[5];
  const float* qkv_b  = (const float*)d_in[6];
  const float* dcw    = (const float*)d_in[7];   // (72,1,3,3) flat
  const float* dcb    = (const float*)d_in[8];
  const float* dc1w   = (const float*)d_in[9];   // (72,1,3,3) flat
  const float* dc1b   = (const float*)d_in[10];
  const float* rpb    = (const float*)d_in[11];  // (NH*9) flat
  const float* proj_w = (const float*)d_in[12];
  const float* proj_b = (const float*)d_in[13];
  float* out = (float*)d_out;

  float* f    = (float*)d_ws;                        // [B][192][HW]
  float* attn = f + (size_t)BN * NQKV * HWSZ;        // [B][64][HW]

  qkv_gemm_kernel<<<dim3(HWSZ / 16, BN), dim3(128), 0, stream>>>(
      x, qkv_w, qkv_b, f);

  conv_attn_kernel<<<dim3(4, 4, BN * NHEAD), dim3(16, 16), 0, stream>>>(
      f, dcw, dcb, dc1w, dc1b, rpb, attn);

  proj_gemm_kernel<<<dim3(HWSZ / 16, BN), dim3(128), 0, stream>>>(
      attn, proj_w, proj_b, out);
}